// ScaledDotProductAttention_17154099380895
// MI455X (gfx1250) — compile-verified
//
#include <hip/hip_runtime.h>

typedef __attribute__((ext_vector_type(2))) float v2f;
typedef __attribute__((ext_vector_type(4))) float v4f;
typedef __attribute__((ext_vector_type(8))) float v8f;

#define SEQ 1024
#define DH  64
#define NB  4          // batch
#define NH  16         // heads
#define NEGV -1000000000.0f

// One wave (32 lanes) handles a 16-query tile for one (b,h).
// grid.x = (SEQ/16) * NB * NH = 64 * 64 = 4096 blocks of 32 threads.
__global__ __launch_bounds__(32) void attn_f32_wmma_kernel(
    const float* __restrict__ Q, const float* __restrict__ Km,
    const float* __restrict__ V, const int*  __restrict__ mask,
    float* __restrict__ outp, float* __restrict__ attnp)
{
    // 16 query rows x 1024 keys of f32 scores = 64 KB LDS
    __shared__ float sm[16 * SEQ];
    __shared__ float mbuf[16];      // per-row max, row-indexed

    const int lane = threadIdx.x;   // 0..31
    const int half = lane >> 4;     // 0 or 1
    const int l16  = lane & 15;

    const int qt = blockIdx.x & 63;   // query tile within sequence
    const int bh = blockIdx.x >> 6;   // fused (b,h), 0..63
    const int b  = bh >> 4;           // NH == 16
    const int q0 = qt * 16;

    const float* Qb = Q  + (size_t)bh * SEQ * DH;
    const float* Kb = Km + (size_t)bh * SEQ * DH;
    const float* Vb = V  + (size_t)bh * SEQ * DH;
    const int*   Mb = mask + (size_t)b * SEQ * SEQ;

    // ---- Load Q tile (16x64) as 16 A-fragments of shape 16x4 ----
    // A layout: M = lane&15 ; lanes 0-15 hold K=(0,1), lanes 16-31 hold K=(2,3)
    v2f qa[16];
    #pragma unroll
    for (int kk = 0; kk < 16; ++kk) {
        const float* p = Qb + (size_t)(q0 + l16) * DH + kk * 4 + half * 2;
        qa[kk] = *(const v2f*)p;     // 8B-aligned -> global_load_b64
    }

    float mrow[8];
    #pragma unroll
    for (int j = 0; j < 8; ++j) mrow[j] = -3.0e38f;

    // ---- Pass 1: S = (Q K^T) * 1/sqrt(64), mask, stash in LDS, track row max ----
    for (int t = 0; t < SEQ / 16; ++t) {
        const int n0 = t * 16;
        // 4 independent accumulator chains to keep the XDL pipe busy
        v8f acc[4];
        #pragma unroll
        for (int u = 0; u < 4; ++u) {
            v8f z = {0.f, 0.f, 0.f, 0.f, 0.f, 0.f, 0.f, 0.f};
            acc[u] = z;
        }
        #pragma unroll
        for (int kk = 0; kk < 16; ++kk) {
            // B layout: N = lane&15 ; lanes 0-15 -> K=(0,1), lanes 16-31 -> K=(2,3)
            // B[k][n] = K[n0+n][k]  (K^T), so read along K's row n0+l16
            const float* p = Kb + (size_t)(n0 + l16) * DH + kk * 4 + half * 2;
            v2f bv = *(const v2f*)p;
            acc[kk & 3] = __builtin_amdgcn_wmma_f32_16x16x4_f32(
                false, qa[kk], false, bv, (short)0, acc[kk & 3], false, false);
        }
        v8f c;
        #pragma unroll
        for (int j = 0; j < 8; ++j)
            c[j] = (acc[0][j] + acc[1][j]) + (acc[2][j] + acc[3][j]);

        // C layout: VGPR j -> M = j + 8*half, N = l16
        #pragma unroll
        for (int j = 0; j < 8; ++j) {
            const int M = j + half * 8;
            const int mk = Mb[(size_t)(q0 + M) * SEQ + n0 + l16];
            const float s = (mk == 0) ? NEGV : c[j] * 0.125f;  // 1/sqrt(64)
            sm[M * SEQ + n0 + l16] = s;
            mrow[j] = fmaxf(mrow[j], s);
        }
    }

    // Row max: reduce across the 16 lanes of each half-wave
    #pragma unroll
    for (int j = 0; j < 8; ++j) {
        #pragma unroll
        for (int off = 1; off < 16; off <<= 1)
            mrow[j] = fmaxf(mrow[j], __shfl_xor(mrow[j], off, 32));
    }
    // Transpose per-row max into row-indexed LDS: lane 0 -> rows 0-7, lane 16 -> rows 8-15
    if (l16 == 0) {
        #pragma unroll
        for (int j = 0; j < 8; ++j) mbuf[half * 8 + j] = mrow[j];
    }
    __syncthreads();   // fences pass-1 scores + mbuf for cross-lane row-major reads
    const float mr = mbuf[l16];   // max of row l16 (same in both halves)

    // ---- Pass 2 (row-major): p = exp(s - m), accumulate row sums ----
    // lane owns row l16, cols cb..cb+7 of each tile: 2x b128 LDS load/store
    float ps = 0.f;
    for (int t = 0; t < SEQ / 16; ++t) {
        float* p = &sm[l16 * SEQ + t * 16 + half * 8];
        v4f a = *(const v4f*)p;
        v4f e = *(const v4f*)(p + 4);
        #pragma unroll
        for (int i = 0; i < 4; ++i) {
            a[i] = __expf(a[i] - mr);
            e[i] = __expf(e[i] - mr);
        }
        *(v4f*)p       = a;
        *(v4f*)(p + 4) = e;
        ps += (a[0] + a[1]) + (a[2] + a[3]) + (e[0] + e[1]) + (e[2] + e[3]);
    }
    // combine the two half-rows (same l16, half 0/1): xor-16 crosses halves
    ps += __shfl_xor(ps, 16, 32);
    const float linv = 1.0f / ps;

    // ---- Pass 3: normalize (row-major), stream attn (b128), O += P V ----
    v8f o[4];
    #pragma unroll
    for (int nb = 0; nb < 4; ++nb) {
        v8f z = {0.f, 0.f, 0.f, 0.f, 0.f, 0.f, 0.f, 0.f};
        o[nb] = z;
    }
    float* attn_bh = attnp + (size_t)bh * SEQ * SEQ;

    for (int t = 0; t < SEQ / 16; ++t) {
        const int n0 = t * 16;
        const int cb = n0 + half * 8;
        // normalize row-major in registers, write back to LDS and to HBM
        float* p = &sm[l16 * SEQ + cb];
        v4f a = *(const v4f*)p * linv;
        v4f e = *(const v4f*)(p + 4) * linv;
        *(v4f*)p       = a;
        *(v4f*)(p + 4) = e;
        float* drow = &attn_bh[(size_t)(q0 + l16) * SEQ + cb];
        *(v4f*)(drow)     = a;     // 16B stores, 32B-aligned cols
        *(v4f*)(drow + 4) = e;
        __syncthreads();   // 1-wave WG: cheap; fences LDS store->load across lanes

        // O(16x64) += P(16x16) * V(16x64): 4 k-chunks x 4 column blocks
        #pragma unroll
        for (int kk = 0; kk < 4; ++kk) {
            const int kb = n0 + kk * 4 + half * 2;   // global key index of K=(0|2)
            v2f av;                                   // A: row M=l16, K pair
            av.x = sm[l16 * SEQ + kb];
            av.y = sm[l16 * SEQ + kb + 1];
            #pragma unroll
            for (int nb = 0; nb < 4; ++nb) {
                v2f bv;                               // B[k][n] = V[key][d0+n]
                bv.x = Vb[(size_t)kb       * DH + nb * 16 + l16];
                bv.y = Vb[(size_t)(kb + 1) * DH + nb * 16 + l16];
                o[nb] = __builtin_amdgcn_wmma_f32_16x16x4_f32(
                    false, av, false, bv, (short)0, o[nb], false, false);
            }
        }
    }

    // ---- Write O (C layout: VGPR j -> M = j + 8*half, N = l16) ----
    #pragma unroll
    for (int nb = 0; nb < 4; ++nb) {
        #pragma unroll
        for (int j = 0; j < 8; ++j) {
            const int M = j + half * 8;
            outp[(size_t)bh * SEQ * DH + (size_t)(q0 + M) * DH + nb * 16 + l16] = o[nb][j];
        }
    }
}

extern "C" void kernel_launch(void* const* d_in, const int* in_sizes, int n_in,
                              void* d_out, int out_size, void* d_ws, size_t ws_size,
                              hipStream_t stream) {
    const float* Q    = (const float*)d_in[0];
    const float* K    = (const float*)d_in[1];
    const float* V    = (const float*)d_in[2];
    const int*   mask = (const int*)d_in[3];

    float* outp  = (float*)d_out;                                   // (B,H,S,D)
    float* attnp = outp + (size_t)NB * NH * SEQ * DH;               // (B,H,S,S)

    dim3 grid((SEQ / 16) * NB * NH);   // 4096
    dim3 block(32);                    // one wave32
    hipLaunchKernelGGL(attn_f32_wmma_kernel, grid, block, 0, stream,
                       Q, K, V, mask, outp, attnp);
}